// SourcePredictor_39032662786525
// MI455X (gfx1250) — compile-verified
//
#include <hip/hip_runtime.h>
#include <math.h>

typedef float f32x2 __attribute__((ext_vector_type(2)));
typedef float f32x8 __attribute__((ext_vector_type(8)));

#define HIDDEN 1024
#define OUT 8192
#define NSUBJ 8
#define BATCH 64
#define TLEN 512
#define MAX_TILES 16
#define TCHUNK 8            // T split for mean-pool parallelism
#define GEMM_THREADS 256    // 8 waves per block -> 2x occupancy per LDS byte

// ---------------------------------------------------------------------------
// 1a) Mean pool, stage 1: partial sums over T chunks.
//     grid.x = B * TCHUNK, 256 threads (one float4 column each).
// ---------------------------------------------------------------------------
__global__ __launch_bounds__(256) void meanpool_partial_kernel(
    const float* __restrict__ ctx, float* __restrict__ part) {
  const int b = blockIdx.x >> 3;          // TCHUNK = 8
  const int c = blockIdx.x & 7;
  const int h4 = threadIdx.x;             // 0..255  (H/4)
  const int t0 = c * (TLEN / TCHUNK);
  const float4* p = reinterpret_cast<const float4*>(ctx) +
                    (size_t)b * (TLEN * (HIDDEN / 4)) +
                    (size_t)t0 * (HIDDEN / 4) + h4;
  float sx = 0.f, sy = 0.f, sz = 0.f, sw = 0.f;
#pragma unroll 4
  for (int t = 0; t < TLEN / TCHUNK; ++t) {
    float4 v = p[(size_t)t * (HIDDEN / 4)];
    sx += v.x; sy += v.y; sz += v.z; sw += v.w;
  }
  float4 o; o.x = sx; o.y = sy; o.z = sz; o.w = sw;
  reinterpret_cast<float4*>(part)[((size_t)b * TCHUNK + c) * (HIDDEN / 4) + h4] = o;
}

// ---------------------------------------------------------------------------
// 1b) Mean pool, stage 2: reduce TCHUNK partials, scale by 1/T.
// ---------------------------------------------------------------------------
__global__ __launch_bounds__(256) void meanpool_reduce_kernel(
    const float* __restrict__ part, float* __restrict__ h0) {
  const int tid = blockIdx.x * blockDim.x + threadIdx.x;  // 0 .. B*(H/4)-1
  const int b = tid >> 8;
  const int h4 = tid & 255;
  const float4* p = reinterpret_cast<const float4*>(part) +
                    (size_t)b * TCHUNK * (HIDDEN / 4) + h4;
  float sx = 0.f, sy = 0.f, sz = 0.f, sw = 0.f;
#pragma unroll
  for (int c = 0; c < TCHUNK; ++c) {
    float4 v = p[(size_t)c * (HIDDEN / 4)];
    sx += v.x; sy += v.y; sz += v.z; sw += v.w;
  }
  const float inv = 1.0f / (float)TLEN;
  float4 o; o.x = sx * inv; o.y = sy * inv; o.z = sz * inv; o.w = sw * inv;
  reinterpret_cast<float4*>(h0)[(size_t)b * (HIDDEN / 4) + h4] = o;
}

// ---------------------------------------------------------------------------
// Shared WMMA tile core: one wave computes a 16x16 f32 output tile by
// iterating K in steps of 4 with V_WMMA_F32_16X16X4_F32.
// A tile lives in LDS, K-major: As[k*16 + m]  (conflict-free ds_load).
// ---------------------------------------------------------------------------
__device__ __forceinline__ f32x8 wmma_tile_k1024(
    const float* __restrict__ As,   // LDS, 1024*16 floats, K-major
    const float* __restrict__ Bw,   // global, row-major K x N
    int n0, int N, int lane) {
  const int col = lane & 15;
  const int hiK = (lane >> 4) << 1;  // 0 for lanes 0-15, 2 for lanes 16-31
  const int m   = lane & 15;
  f32x8 acc = {};
  for (int k0 = 0; k0 < HIDDEN; k0 += 4) {
    f32x2 a, b;
    a.x = As[(k0 + hiK) * 16 + m];
    a.y = As[(k0 + hiK + 1) * 16 + m];
    const float* bp = Bw + (size_t)(k0 + hiK) * N + n0 + col;
    b.x = bp[0];
    b.y = bp[N];
    acc = __builtin_amdgcn_wmma_f32_16x16x4_f32(
        false, a, false, b, (short)0, acc, false, false);
  }
  return acc;
}

// ---------------------------------------------------------------------------
// 2) Dense WMMA GEMM: C[M,N] = op(A[M,1024] @ B[1024,N] + bias)
//    OP: 0 = exact GELU, 1 = none, 2 = clamp[-10,2]
//    Block = 256 threads (8 waves) sharing one 16-row M tile (64KB LDS),
//    each wave computes one 16-col N tile.
// ---------------------------------------------------------------------------
template <int OP>
__global__ __launch_bounds__(GEMM_THREADS) void wmma_gemm_dense(
    const float* __restrict__ A, const float* __restrict__ Bw,
    const float* __restrict__ bias, float* __restrict__ C, int N) {
  __shared__ float As[HIDDEN * 16];  // K-major, 64 KB
  const int m0 = blockIdx.y * 16;
  for (int idx = threadIdx.x; idx < 16 * HIDDEN; idx += GEMM_THREADS) {
    int m = idx >> 10, k = idx & (HIDDEN - 1);          // global-coalesced read
    As[k * 16 + m] = A[(size_t)(m0 + m) * HIDDEN + k];
  }
  __syncthreads();

  const int lane = threadIdx.x & 31;
  const int wave = threadIdx.x >> 5;
  const int n0 = (blockIdx.x * 8 + wave) * 16;
  f32x8 acc = wmma_tile_k1024(As, Bw, n0, N, lane);

  const int col = lane & 15;
  const int rofs = (lane >> 4) << 3;
  const float bn = bias[n0 + col];
#pragma unroll
  for (int r = 0; r < 8; ++r) {
    float v = acc[r] + bn;
    if (OP == 0) v = 0.5f * v * (1.0f + erff(v * 0.70710678118654752f));
    if (OP == 2) v = fminf(2.0f, fmaxf(-10.0f, v));
    C[(size_t)(m0 + r + rofs) * N + n0 + col] = v;
  }
}

// ---------------------------------------------------------------------------
// 3) Subject tile setup (device-side grouping; B=64, trivial serial work).
//    meta[0] = numTiles; per tile: [subject, rows[16] (-1 pad)], stride 17.
// ---------------------------------------------------------------------------
__global__ void subj_setup_kernel(const int* __restrict__ sid,
                                  int* __restrict__ meta) {
  if (threadIdx.x != 0 || blockIdx.x != 0) return;
  int nt = 0;
  for (int s = 0; s < NSUBJ; ++s) {
    int rows[BATCH]; int c = 0;
    for (int b = 0; b < BATCH; ++b)
      if (sid[b] == s) rows[c++] = b;
    for (int off = 0; off < c; off += 16) {
      int* rec = meta + 1 + nt * 17;
      rec[0] = s;
      for (int i = 0; i < 16; ++i)
        rec[1 + i] = (off + i < c) ? rows[off + i] : -1;
      ++nt;
    }
  }
  meta[0] = nt;
}

// ---------------------------------------------------------------------------
// 4) Grouped subject head: mu = gather(h) @ subj_w[s] + subj_b[s], plus fused
//    x0 = mu + eps * exp(0.5 * log_var)   (log_var already in d_out).
// ---------------------------------------------------------------------------
__global__ __launch_bounds__(GEMM_THREADS) void wmma_head_mu(
    const float* __restrict__ H2, const float* __restrict__ SW,
    const float* __restrict__ SB, const float* __restrict__ eps,
    const float* __restrict__ logv, const int* __restrict__ meta,
    float* __restrict__ mu, float* __restrict__ x0) {
  const int nt = meta[0];
  if ((int)blockIdx.y >= nt) return;  // uniform per block

  __shared__ int rows[16];
  __shared__ float As[HIDDEN * 16];  // K-major gathered A tile, 64 KB
  const int* rec = meta + 1 + blockIdx.y * 17;
  const int s = rec[0];
  if (threadIdx.x < 16) rows[threadIdx.x] = rec[1 + threadIdx.x];
  __syncthreads();
  for (int idx = threadIdx.x; idx < 16 * HIDDEN; idx += GEMM_THREADS) {
    int m = idx >> 10, k = idx & (HIDDEN - 1);
    int r = rows[m];
    As[k * 16 + m] = (r >= 0) ? H2[(size_t)r * HIDDEN + k] : 0.0f;
  }
  __syncthreads();

  const int lane = threadIdx.x & 31;
  const int wave = threadIdx.x >> 5;
  const int n0 = (blockIdx.x * 8 + wave) * 16;
  const float* Bw = SW + ((size_t)s * HIDDEN * OUT);
  f32x8 acc = wmma_tile_k1024(As, Bw, n0, OUT, lane);

  const int col = lane & 15;
  const int rofs = (lane >> 4) << 3;
  const float bn = SB[(size_t)s * OUT + n0 + col];
#pragma unroll
  for (int r = 0; r < 8; ++r) {
    int bidx = rows[r + rofs];
    if (bidx >= 0) {
      size_t o = (size_t)bidx * OUT + n0 + col;
      float m_ = acc[r] + bn;
      mu[o] = m_;
      x0[o] = m_ + eps[o] * expf(0.5f * logv[o]);
    }
  }
}

// ---------------------------------------------------------------------------
extern "C" void kernel_launch(void* const* d_in, const int* in_sizes, int n_in,
                              void* d_out, int out_size, void* d_ws,
                              size_t ws_size, hipStream_t stream) {
  const float* ctx = (const float*)d_in[0];
  const int*   sid = (const int*)d_in[1];
  const float* eps = (const float*)d_in[2];
  const float* w1  = (const float*)d_in[3];
  const float* b1  = (const float*)d_in[4];
  const float* w2  = (const float*)d_in[5];
  const float* b2  = (const float*)d_in[6];
  const float* sw  = (const float*)d_in[7];
  const float* sb  = (const float*)d_in[8];
  const float* lvw = (const float*)d_in[9];
  const float* lvb = (const float*)d_in[10];

  float* out = (float*)d_out;
  float* x0  = out;                         // (B, OUT)
  float* mu  = out + (size_t)BATCH * OUT;   // (B, OUT)
  float* lv  = out + (size_t)2 * BATCH * OUT;

  // Workspace layout. `part` (2 MB) is dead after the reduce kernel, so h1/h2
  // alias its space (stream-ordered writes happen strictly after last read).
  float* h0   = (float*)d_ws;                      // (B, H) pooled
  float* part = h0 + BATCH * HIDDEN;               // (B, TCHUNK, H) partials
  float* h1   = part;                              // (B, H) after GELU layer
  float* h2   = h1 + BATCH * HIDDEN;               // (B, H) final latent
  int*   meta = (int*)(part + (size_t)BATCH * TCHUNK * HIDDEN);

  // 1) mean pool: split-T partials (4096 waves) then reduce
  meanpool_partial_kernel<<<dim3(BATCH * TCHUNK), 256, 0, stream>>>(ctx, part);
  meanpool_reduce_kernel<<<dim3((BATCH * HIDDEN / 4) / 256), 256, 0, stream>>>(
      part, h0);
  // 2) MLP layer 1 (+ exact GELU)
  wmma_gemm_dense<0><<<dim3(HIDDEN / 128, BATCH / 16), GEMM_THREADS, 0, stream>>>(
      h0, w1, b1, h1, HIDDEN);
  // 3) MLP layer 2
  wmma_gemm_dense<1><<<dim3(HIDDEN / 128, BATCH / 16), GEMM_THREADS, 0, stream>>>(
      h1, w2, b2, h2, HIDDEN);
  // 4) log-var head (+ clamp) -> d_out slice
  wmma_gemm_dense<2><<<dim3(OUT / 128, BATCH / 16), GEMM_THREADS, 0, stream>>>(
      h2, lvw, lvb, lv, OUT);
  // 5) device-side subject grouping
  subj_setup_kernel<<<1, 32, 0, stream>>>(sid, meta);
  // 6) grouped subject head + fused reparameterized sample
  wmma_head_mu<<<dim3(OUT / 128, MAX_TILES), GEMM_THREADS, 0, stream>>>(
      h2, sw, sb, eps, lv, meta, mu, x0);
}